// VectorQuantizerEMA_3281355014182
// MI455X (gfx1250) — compile-verified
//
#include <hip/hip_runtime.h>

// Problem geometry (from reference):
//   z: (32, 64, 32, 32) f32 ; zf: (32768, 64) ; embedding: (1024, 64)
//   out = z_q (2097152 f32) ++ loss (1 f32) ++ counts (1024 f32)
#define NB      32
#define EDIM    64
#define HW      1024        // 32*32
#define NROWS   32768       // NB*HW
#define NE      1024
#define NELEM   2097152     // NB*EDIM*HW

// LDS staging for the argmin kernel: 512 codebook rows per pass,
// pitch 68 dwords (conflict-free B-fragment reads), plus 512 norms.
#define STAGE_ROWS   512
#define EMB_PITCH    68
#define LDS_FLOATS   (STAGE_ROWS * EMB_PITCH + STAGE_ROWS)
#define LDS_BYTES    (LDS_FLOATS * 4)

typedef float v2f __attribute__((ext_vector_type(2)));
typedef float v8f __attribute__((ext_vector_type(8)));

#if defined(__has_builtin)
# if __has_builtin(__builtin_amdgcn_global_load_async_to_lds_b128) && \
     __has_builtin(__builtin_amdgcn_s_wait_asynccnt)
#  define HAVE_ASYNC_LDS 1
# endif
#endif
#ifndef HAVE_ASYNC_LDS
# define HAVE_ASYNC_LDS 0
#endif

// Types matching the async-LDS builtin prototype (from hipcc's diagnostic):
//   parameter type is 'int __attribute__((vector_size(4*sizeof(int)))) *'
typedef int v4i_g __attribute__((vector_size(4 * sizeof(int))));
typedef __attribute__((address_space(1))) v4i_g glb_v4i;
typedef __attribute__((address_space(3))) v4i_g lds_v4i;

// ---------------------------------------------------------------------------
// Kernel 0: zero the workspace accumulators + loss slot (ws is poisoned 0xAA).
// ---------------------------------------------------------------------------
__global__ void vq_init(float* __restrict__ counts, float* __restrict__ sums,
                        float* __restrict__ lossOut) {
    int t = blockIdx.x * blockDim.x + threadIdx.x;   // 65536 threads
    if (t < NE) counts[t] = 0.0f;
    if (t < NE * EDIM) sums[t] = 0.0f;
    if (t == 0) *lossOut = 0.0f;
}

// ---------------------------------------------------------------------------
// Kernel 1: fused distance GEMM + per-row argmin using V_WMMA_F32_16X16X4_F32.
// 512-thread workgroup = 16 waves, one 16-row tile per wave (256 rows/WG).
// Codebook staged into LDS in 2 passes of 512 rows; B fragments and ||e||^2
// served from LDS (conflict-free with pitch 68). argmin target:
//   ||e||^2 - 2 * (zf . e)     (||z||^2 is row-constant, drops out of argmin)
// ---------------------------------------------------------------------------
__global__ __launch_bounds__(512)
void vq_argmin(const float* __restrict__ z, const float* __restrict__ emb,
               int* __restrict__ idxOut) {
    extern __shared__ float ldsBuf[];
    float* ldsEmb   = ldsBuf;                          // 512 x 68 dwords
    float* eNormLds = ldsBuf + STAGE_ROWS * EMB_PITCH; // 512 dwords

    const int tid  = threadIdx.x;        // 0..511
    const int lane = tid & 31;
    const int wave = tid >> 5;           // 0..15
    const int m    = lane & 15;          // row / col within 16
    const int half = lane >> 4;          // 0 or 1

    const int rowBase = blockIdx.x * 256 + wave * 16; // 128 blocks
    const int n  = rowBase + m;
    const int b  = n >> 10;              // batch
    const int hw = n & 1023;
    // zf[n][c] = z[b*65536 + c*1024 + hw]
    const float* zb = z + (size_t)b * (EDIM * HW) + hw;

    // A-matrix registers, ISA 16x4 f32 layout:
    //   lanes 0-15 : VGPR0=K0, VGPR1=K1 ; lanes 16-31 : VGPR0=K2, VGPR1=K3
    v2f a[16];
#pragma unroll
    for (int kk = 0; kk < 16; ++kk) {
        const int k0 = 4 * kk + 2 * half;
        a[kk].x = zb[(size_t)(k0    ) * HW];
        a[kk].y = zb[(size_t)(k0 + 1) * HW];
    }

    float minVal[8];
    int   minIdx[8];
#pragma unroll
    for (int v = 0; v < 8; ++v) { minVal[v] = __builtin_inff(); minIdx[v] = 0; }

    for (int pass = 0; pass < 2; ++pass) {
        // ---- Stage 512 codebook rows (128 KB) into LDS, coalesced b128s ----
        const float* src0 = emb + (size_t)pass * STAGE_ROWS * EDIM;
#pragma unroll
        for (int i = 0; i < 16; ++i) {               // 8192 float4s / 512 thr
            const int g    = i * 512 + tid;          // float4 index
            const int rowl = g >> 4;                 // 16 float4s per row
            const int c4   = g & 15;
            const float* src = src0 + (size_t)g * 4;
            float*       dst = ldsEmb + rowl * EMB_PITCH + c4 * 4;
#if HAVE_ASYNC_LDS
            __builtin_amdgcn_global_load_async_to_lds_b128(
                (glb_v4i*)src, (lds_v4i*)dst, 0, 0);
#else
            *(float4*)dst = *(const float4*)src;
#endif
        }
#if HAVE_ASYNC_LDS
        __builtin_amdgcn_s_wait_asynccnt(0);
#endif
        __syncthreads();

        // ---- Precompute ||e||^2 for the staged rows (1 row / thread) ----
        {
            const float* r = ldsEmb + tid * EMB_PITCH;
            float s = 0.0f;
#pragma unroll
            for (int j = 0; j < 16; ++j) {
                float4 q = *(const float4*)(r + j * 4);
                s += q.x * q.x + q.y * q.y + q.z * q.z + q.w * q.w;
            }
            eNormLds[tid] = s;
        }
        __syncthreads();

        // ---- 32 column tiles of 16 codes, B + norms from LDS ----
        for (int ct = 0; ct < 32; ++ct) {
            const int eL = ct * 16 + m;              // staged-local code row
            const int e  = pass * STAGE_ROWS + eL;   // global code index
            const float* er = ldsEmb + eL * EMB_PITCH + 2 * half;

            v2f bm[16];                              // ds_load_b64, bank-free
#pragma unroll
            for (int kk = 0; kk < 16; ++kk) bm[kk] = *(const v2f*)(er + 4 * kk);
            const float pn = eNormLds[eL];

            // 16 chained fp32 WMMAs: K = 64
            v8f acc = {0.f, 0.f, 0.f, 0.f, 0.f, 0.f, 0.f, 0.f};
#pragma unroll
            for (int kk = 0; kk < 16; ++kk)
                acc = __builtin_amdgcn_wmma_f32_16x16x4_f32(
                    false, a[kk], false, bm[kk], (short)0, acc, false, false);

            // D layout: lane holds col n=m, rows 8*half + v (v = VGPR index)
#pragma unroll
            for (int v = 0; v < 8; ++v) {
                const float dist = pn - 2.0f * acc[v];
                if (dist < minVal[v]) { minVal[v] = dist; minIdx[v] = e; }
            }
        }
        __syncthreads();   // before next pass overwrites the staged tile
    }

    // Reduce across the 16 lanes of each half (same rows, different cols).
#pragma unroll
    for (int off = 1; off < 16; off <<= 1) {
#pragma unroll
        for (int v = 0; v < 8; ++v) {
            const float oV = __shfl_xor(minVal[v], off, 32);
            const int   oI = __shfl_xor(minIdx[v], off, 32);
            if (oV < minVal[v] || (oV == minVal[v] && oI < minIdx[v])) {
                minVal[v] = oV; minIdx[v] = oI;
            }
        }
    }
    if (m == 0) {
#pragma unroll
        for (int v = 0; v < 8; ++v)
            idxOut[rowBase + 8 * half + v] = minIdx[v];
    }
}

// ---------------------------------------------------------------------------
// Kernel 2: segment sums (counts, per-dim sums) via fp32 global atomics.
// Thread t -> (b, c, hw) so z reads are fully coalesced along hw.
// ---------------------------------------------------------------------------
__global__ __launch_bounds__(256)
void vq_scatter(const float* __restrict__ z, const int* __restrict__ idxArr,
                float* __restrict__ counts, float* __restrict__ sums) {
    const int t  = blockIdx.x * blockDim.x + threadIdx.x;   // 0..NELEM-1
    const int hw = t & 1023;
    const int c  = (t >> 10) & 63;
    const int b  = t >> 16;
    const int n  = b * HW + hw;
    const int e  = idxArr[n];
    unsafeAtomicAdd(&sums[e * EDIM + c], z[t]);
    if (c == 0) unsafeAtomicAdd(&counts[e], 1.0f);
}

// ---------------------------------------------------------------------------
// Kernel 3: EMA buffer update + refreshed embedding; emit raw counts output.
// ---------------------------------------------------------------------------
__global__ __launch_bounds__(256)
void vq_ema(const float* __restrict__ clusterSize, const float* __restrict__ clusterSum,
            const float* __restrict__ counts, const float* __restrict__ sums,
            float* __restrict__ newEmbed, float* __restrict__ outCounts) {
    const int t = blockIdx.x * blockDim.x + threadIdx.x;    // 0..65535
    const int e = t >> 6;
    const int c = t & 63;
    const float ncs  = 0.99f * clusterSize[e] + 0.01f * counts[e];
    const float nsum = 0.99f * clusterSum[t]  + 0.01f * sums[t];
    newEmbed[t] = nsum / (ncs + 1e-5f);
    if (c == 0) outCounts[e] = counts[e];
}

// ---------------------------------------------------------------------------
// Kernel 4: gather-dequantize z_q (already in (B,C,H,W) layout) + commitment
// loss: BETA * mean((z_q - z)^2), block-reduced in LDS then atomically added.
// ---------------------------------------------------------------------------
__global__ __launch_bounds__(256)
void vq_gather(const float* __restrict__ z, const int* __restrict__ idxArr,
               const float* __restrict__ newEmbed, float* __restrict__ out,
               float* __restrict__ lossOut) {
    const int t  = blockIdx.x * blockDim.x + threadIdx.x;
    const int hw = t & 1023;
    const int c  = (t >> 10) & 63;
    const int b  = t >> 16;
    const int n  = b * HW + hw;
    const int e  = idxArr[n];
    const float zq = newEmbed[e * EDIM + c];
    const float zv = z[t];
    out[t] = zq;                               // straight-through forward == z_q
    const float d = zq - zv;

    __shared__ float red[256];
    red[threadIdx.x] = d * d;
    __syncthreads();
    for (int s = 128; s > 0; s >>= 1) {
        if (threadIdx.x < s) red[threadIdx.x] += red[threadIdx.x + s];
        __syncthreads();
    }
    if (threadIdx.x == 0)
        unsafeAtomicAdd(lossOut, red[0] * (0.25f / (float)NELEM));
}

// ---------------------------------------------------------------------------
extern "C" void kernel_launch(void* const* d_in, const int* in_sizes, int n_in,
                              void* d_out, int out_size, void* d_ws, size_t ws_size,
                              hipStream_t stream) {
    const float* z     = (const float*)d_in[0];   // (32,64,32,32)
    const float* emb   = (const float*)d_in[1];   // (1024,64)
    const float* csize = (const float*)d_in[2];   // (1024,)
    const float* csum  = (const float*)d_in[3];   // (1024,64)

    float* out       = (float*)d_out;
    float* lossOut   = out + NELEM;               // scalar
    float* countsOut = out + NELEM + 1;           // (1024,)

    // Workspace layout (floats/ints, ~659 KB total):
    int*   idxArr   = (int*)d_ws;                               // 32768 ints
    float* wsf      = (float*)d_ws;
    float* counts   = wsf + NROWS;                              // 1024
    float* sums     = wsf + NROWS + NE;                         // 65536
    float* newEmbed = wsf + NROWS + NE + NE * EDIM;             // 65536

    // Allow >64KB dynamic LDS for the staged-codebook argmin kernel.
    (void)hipFuncSetAttribute((const void*)vq_argmin,
                              hipFuncAttributeMaxDynamicSharedMemorySize,
                              LDS_BYTES);

    vq_init   <<<256, 256, 0, stream>>>(counts, sums, lossOut);
    vq_argmin <<<NROWS / 256, 512, LDS_BYTES, stream>>>(z, emb, idxArr);
    vq_scatter<<<NELEM / 256, 256, 0, stream>>>(z, idxArr, counts, sums);
    vq_ema    <<<(NE * EDIM) / 256, 256, 0, stream>>>(csize, csum, counts, sums,
                                                      newEmbed, countsOut);
    vq_gather <<<NELEM / 256, 256, 0, stream>>>(z, idxArr, newEmbed, out, lossOut);
}